// ICUSTOMIntSoftmax_65687229826223
// MI455X (gfx1250) — compile-verified
//
#include <hip/hip_runtime.h>
#include <math.h>
#include <stdint.h>

// ---------------------------------------------------------------------------
// Integer softmax (I-BERT style) over rows of 1024 f32 elements.
//
// MI455X reasoning: 402 MB of HBM traffic -> ~17 us floor at 23.3 TB/s.
// Memory-bound; one wave32 per row, row staged into LDS once via the gfx1250
// async global->LDS path (ASYNCcnt), then three LDS passes (max / poly+sum /
// normalize+store). Reductions are intra-wave shfl_xor butterflies (wave32).
// ---------------------------------------------------------------------------

#define ROW_LEN          1024
#define WAVES_PER_BLOCK  8
#define N_SEG            16
// ws float layout: [0]=s ; [1..17]=lo_i[0..16] (entry 16 = +big sentinel) ;
// [20+4k+0..2] = int coeffs (c2,c1,c0 scaled), [20+4k+3] = 0   -> 84 floats
#define WS_FLOATS        84

// ---------------------------------------------------------------------------
// Setup kernel: replicate np.polyfit (least squares, degree 2) of exp(x) on
// each of the 16 segments of [-11,0], over the same 10000-point float32 grid,
// then apply the runtime-s-dependent integer scaling with floor().
// ---------------------------------------------------------------------------
__global__ void int_softmax_setup(const float* __restrict__ sf,
                                  float* __restrict__ ws,
                                  float* __restrict__ out_sf_ptr) {
  const int k = threadIdx.x;
  const float s = sf[0];
  if (k == 0) ws[0] = s;
  if (k == 31 && out_sf_ptr) *out_sf_ptr = 1.0f / 256.0f;  // second tuple output
  if (k < 17) {
    if (k == 16) {
      ws[1 + 16] = 3.0e38f;  // sentinel: no segment above 15
    } else {
      float lo = (float)(-11.0 + (11.0 / 16.0) * (double)k);
      ws[1 + k] = floorf(lo / s);
    }
  }
  if (k >= N_SEG) return;

  const double lo = -11.0 + (11.0 / 16.0) * (double)k;
  const double hi = lo + 11.0 / 16.0;
  const float lof = (float)lo, hif = (float)hi;

  double S0 = 0, S1 = 0, S2 = 0, S3 = 0, S4 = 0, T0 = 0, T1 = 0, T2 = 0;
  for (int i = 0; i < 10000; ++i) {
    float xf = (float)(-11.0 + 11.0 * ((double)i / 9999.0));
    if (xf < lof || xf > hif) continue;
    double x = (double)xf;
    double y = exp(x);
    double x2 = x * x;
    S0 += 1.0; S1 += x; S2 += x2; S3 += x2 * x; S4 += x2 * x2;
    T0 += y;   T1 += x * y; T2 += x2 * y;
  }
  // Normal equations  [[S4,S3,S2],[S3,S2,S1],[S2,S1,S0]] [a,b,c]^T = [T2,T1,T0]
  double det  = S4 * (S2 * S0 - S1 * S1) - S3 * (S3 * S0 - S1 * S2) + S2 * (S3 * S1 - S2 * S2);
  double da   = T2 * (S2 * S0 - S1 * S1) - S3 * (T1 * S0 - T0 * S1) + S2 * (T1 * S1 - T0 * S2);
  double db   = S4 * (T1 * S0 - T0 * S1) - T2 * (S3 * S0 - S1 * S2) + S2 * (S3 * T0 - S2 * T1);
  double dc   = S4 * (S2 * T0 - T1 * S1) - S3 * (S3 * T0 - T1 * S2) + T2 * (S3 * S1 - S2 * S2);
  double a = da / det, b = db / det, c = dc / det;

  const double sd = (double)s;
  ws[20 + 4 * k + 0] = floorf((float)(a * sd * sd * 1048576.0));  // * s^2 * 2^20
  ws[20 + 4 * k + 1] = floorf((float)(b * sd * 1048576.0));       // * s^1 * 2^20
  ws[20 + 4 * k + 2] = floorf((float)(c * 1048576.0));            // * s^0 * 2^20
  ws[20 + 4 * k + 3] = 0.0f;
}

// ---------------------------------------------------------------------------
// Per-element segment evaluation: arithmetic segment-index estimate from the
// uniform float boundaries, then +-1 correction against the exact floored
// integer boundaries (reference: later segments win on boundary overlap).
// ---------------------------------------------------------------------------
__device__ __forceinline__ float seg_eval(float xi, const float* __restrict__ cst,
                                          float s, float inv_esc) {
  float f = (xi * s + 11.0f) * (16.0f / 11.0f);
  int idx = (int)floorf(f);
  idx = idx < 0 ? 0 : (idx > 15 ? 15 : idx);
  const float lo_c = cst[1 + idx];
  const float lo_n = cst[2 + idx];           // entry 16 is +big sentinel
  idx += (xi >= lo_n) ? 1 : 0;               // mutually exclusive corrections
  idx -= (xi <  lo_c) ? 1 : 0;
  idx = idx < 0 ? 0 : (idx > 15 ? 15 : idx);
  const float4 cf = *(const float4*)&cst[20 + 4 * idx];  // ds_load_b128
  float r = fmaf(cf.x, xi, cf.y);
  r = fmaf(r, xi, cf.z);
  float ef = fmaxf(r * (1.0f / 1048576.0f), 1e-10f);     // exp_float, clipped
  return floorf(ef * inv_esc);                           // exp_q
}

__device__ __forceinline__ float wave_max32(float v) {
#pragma unroll
  for (int sh = 16; sh >= 1; sh >>= 1) v = fmaxf(v, __shfl_xor(v, sh, 32));
  return v;
}
__device__ __forceinline__ float wave_sum32(float v) {
#pragma unroll
  for (int sh = 16; sh >= 1; sh >>= 1) v += __shfl_xor(v, sh, 32);
  return v;
}

// ---------------------------------------------------------------------------
// Main kernel: one wave per row, row staged in LDS via async loads.
// ---------------------------------------------------------------------------
__global__ __launch_bounds__(256) void int_softmax_main(
    const float* __restrict__ x, float* __restrict__ out,
    const float* __restrict__ ws, int rows) {
  __shared__ __align__(16) float  cst[96];
  __shared__ __align__(16) float4 tile4[WAVES_PER_BLOCK][ROW_LEN / 4];

  for (int i = threadIdx.x; i < WS_FLOATS; i += blockDim.x) cst[i] = ws[i];
  __syncthreads();

  const int wave = threadIdx.x >> 5;
  const int lane = threadIdx.x & 31;
  const int row  = blockIdx.x * WAVES_PER_BLOCK + wave;
  if (row >= rows) return;  // wave-uniform; EXEC stays all-ones for async ops

  const float s       = cst[0];
  const float inv_s   = 1.0f / s;
  const float inv_esc = 1048576.0f * inv_s;  // 1 / (s / 2^20)

  const float*  xrow = x + (size_t)row * ROW_LEN;
  float*        orow = out + (size_t)row * ROW_LEN;
  float4*       trow = &tile4[wave][0];

  // ---- Stage the 4 KB row into LDS with gfx1250 async global->LDS loads ----
#pragma unroll
  for (int j = 0; j < 8; ++j) {
    const int i = j * 32 + lane;
    unsigned int       ldsoff = (unsigned int)(uintptr_t)(&trow[i]);  // low 32b = LDS addr
    unsigned long long ga     = (unsigned long long)(uintptr_t)(xrow + 4 * i);
    asm volatile("global_load_async_to_lds_b128 %0, %1, off"
                 :: "v"(ldsoff), "v"(ga) : "memory");
  }
  asm volatile("s_wait_asynccnt 0" ::: "memory");

  // ---- Pass 1: row max of floor(x/s) ----
  float m = -3.4e38f;
#pragma unroll
  for (int j = 0; j < 8; ++j) {
    float4 v = trow[j * 32 + lane];
    m = fmaxf(m, floorf(v.x * inv_s));
    m = fmaxf(m, floorf(v.y * inv_s));
    m = fmaxf(m, floorf(v.z * inv_s));
    m = fmaxf(m, floorf(v.w * inv_s));
  }
  m = wave_max32(m);

  // ---- Pass 2: polynomial exp_q + row sum; overwrite tile with exp_q ----
  float acc = 0.0f;
#pragma unroll
  for (int j = 0; j < 8; ++j) {
    float4 v = trow[j * 32 + lane];
    float4 q;
    q.x = seg_eval(floorf(v.x * inv_s) - m, cst, s, inv_esc);
    q.y = seg_eval(floorf(v.y * inv_s) - m, cst, s, inv_esc);
    q.z = seg_eval(floorf(v.z * inv_s) - m, cst, s, inv_esc);
    q.w = seg_eval(floorf(v.w * inv_s) - m, cst, s, inv_esc);
    acc += (q.x + q.y) + (q.z + q.w);
    trow[j * 32 + lane] = q;
  }
  float S = fmaxf(wave_sum32(acc), 1.0f);
  const float factor = floorf(1073741823.0f / S);  // floor(max_val / exp_sum)

  // ---- Pass 3: normalize + streaming store ----
  const float k23 = 1.0f / 8388608.0f;  // 2^-(31-8)
  const float osf = 1.0f / 256.0f;      // 2^-OUTPUT_BIT
#pragma unroll
  for (int j = 0; j < 8; ++j) {
    float4 q = trow[j * 32 + lane];
    float4 o;
    o.x = floorf(q.x * factor * k23) * osf;
    o.y = floorf(q.y * factor * k23) * osf;
    o.z = floorf(q.z * factor * k23) * osf;
    o.w = floorf(q.w * factor * k23) * osf;
    ((float4*)orow)[j * 32 + lane] = o;
  }
}

// ---------------------------------------------------------------------------
extern "C" void kernel_launch(void* const* d_in, const int* in_sizes, int n_in,
                              void* d_out, int out_size, void* d_ws, size_t ws_size,
                              hipStream_t stream) {
  const float* x  = (const float*)d_in[0];
  const float* sf = (const float*)d_in[1];
  float* out = (float*)d_out;
  float* ws  = (float*)d_ws;

  const long long total = (long long)in_sizes[0];     // 4*12*1024*1024
  const int rows = (int)(total / ROW_LEN);
  float* out_sf_ptr = (out_size > (int)total) ? (out + total) : nullptr;

  int_softmax_setup<<<dim3(1), dim3(32), 0, stream>>>(sf, ws, out_sf_ptr);

  const int blocks = (rows + WAVES_PER_BLOCK - 1) / WAVES_PER_BLOCK;
  int_softmax_main<<<dim3(blocks), dim3(256), 0, stream>>>(x, out, ws, rows);
}